// DirectionalAttn_54984171323556
// MI455X (gfx1250) — compile-verified
//
#include <hip/hip_runtime.h>
#include <hip/hip_bf16.h>
#include <stdint.h>

// CDNA5 / gfx1250, wave32. fp32 WMMA (16x16x4) for all dense projections.
typedef __attribute__((ext_vector_type(2))) float v2f;
typedef __attribute__((ext_vector_type(8))) float v8f;

#define SEQ 384
#define DIM 256
#define ITILE 8  // query rows per attention block

enum EpiMode { EPI_NONE = 0, EPI_BIAS_ELU = 1 };

__device__ __forceinline__ float tanh_fast(float x) {
#if __has_builtin(__builtin_amdgcn_tanhf)
  return __builtin_amdgcn_tanhf(x);  // native V_TANH_F32 on gfx1250
#else
  return 1.0f - 2.0f / (__expf(2.0f * x) + 1.0f);
#endif
}

// ---------------------------------------------------------------------------
// Generic fp32 WMMA GEMM: OUT[M,DIM] = epilogue(A[M,DIM] @ W[DIM,DIM])
// Block = 256 threads = 8 waves; wave w computes rows [mBase+16w, +16),
// block column = 16 output features. W panel (DIM x 16 = 16 KB) staged in LDS.
//
// fp32 A-frag layout (ISA 7.12.2, 16x4): lane l -> M = l&15, K = k + (l>>4)*2
// fp32 B-frag layout (4x16): lane l -> N = l&15, rows K = k + (l>>4)*2, +1
// C/D (16x16 f32): VGPR r -> M = r + (l>>4)*8, N = l&15
// ---------------------------------------------------------------------------
template <int MODE>
__global__ __launch_bounds__(256) void gemm_wmma_kernel(
    const float* __restrict__ A, const float* __restrict__ W,
    const float* __restrict__ bias, float* __restrict__ out) {
  __shared__ float sW[DIM * 16];
  const int lane  = threadIdx.x & 31;
  const int wave  = threadIdx.x >> 5;
  const int nBase = blockIdx.y * 16;
  const int mBase = blockIdx.x * 128 + wave * 16;

  for (int idx = threadIdx.x; idx < DIM * 16; idx += 256) {
    const int r = idx >> 4, c = idx & 15;
    sW[idx] = W[r * DIM + nBase + c];
  }
  __syncthreads();

  const int m     = mBase + (lane & 15);
  const int khalf = (lane >> 4) << 1;
  const int n     = lane & 15;

  v8f acc = {};
#pragma unroll 4
  for (int k = 0; k < DIM; k += 4) {
    const int kb = k + khalf;
    v2f a = *(const v2f*)(A + (size_t)m * DIM + kb);  // 8B aligned (kb even)
    v2f b;
    b.x = sW[kb * 16 + n];
    b.y = sW[(kb + 1) * 16 + n];
    acc = __builtin_amdgcn_wmma_f32_16x16x4_f32(false, a, false, b, (short)0,
                                                acc, false, false);
  }

  const int mRow = mBase + (lane >> 4) * 8;
  const int col  = nBase + n;
#pragma unroll
  for (int r = 0; r < 8; ++r) {
    float v = acc[r];
    if (MODE == EPI_BIAS_ELU) {
      v += bias[col];
      v = (v > 0.0f) ? v : (__expf(v) - 1.0f);  // jax.nn.elu, alpha=1
    }
    out[(size_t)(mRow + r) * DIM + col] = v;
  }
}

// ---------------------------------------------------------------------------
// Fused directional attention, 8 query rows per block:
//   attn[b,i,d] = sum_{j<i, mask[b,j]} e_ij * rep[b,j,d] / sum_j e_ij,
//   e_ij = exp(5*tanh((dep[b,j,d] + head[b,i,d] + bias[d]) / 5))
// Logits bounded in [-5,5] -> max-free softmax is exact vs reference; the
// i with no valid j (ssum==0) yields 0, matching the post-softmax re-mask.
// dep/rep row elements are loaded once and reused across ITILE accumulators
// (8x less L2 traffic); inner loop = 1 v_tanh + 1 v_exp + 2 FMA per (i,j,d).
// ---------------------------------------------------------------------------
__global__ __launch_bounds__(DIM) void attn_softmax_kernel(
    const float* __restrict__ rep, const float* __restrict__ dep,
    const float* __restrict__ head, const float* __restrict__ attn_bias,
    const uint8_t* __restrict__ rep_mask, float* __restrict__ out) {
  const int iBase = blockIdx.x * ITILE;
  const int b     = blockIdx.y;
  const int d     = threadIdx.x;

  const float bia = attn_bias[d];
  float h[ITILE], ssum[ITILE], acc[ITILE];
#pragma unroll
  for (int ii = 0; ii < ITILE; ++ii) {
    h[ii]    = head[((size_t)(b * SEQ + iBase + ii)) * DIM + d] + bia;
    ssum[ii] = 0.0f;
    acc[ii]  = 0.0f;
  }

  const size_t colBase = (size_t)b * SEQ * DIM + d;

  // Main body: j < iBase is valid for every i in the tile.
  for (int j = 0; j < iBase; ++j) {
    if (!rep_mask[b * SEQ + j]) continue;  // block-uniform branch
    const float dj = dep[colBase + (size_t)j * DIM];
    const float rj = rep[colBase + (size_t)j * DIM];
#pragma unroll
    for (int ii = 0; ii < ITILE; ++ii) {
      const float t = tanh_fast(0.2f * (dj + h[ii]));
      const float e = __expf(5.0f * t);
      ssum[ii] += e;
      acc[ii] = fmaf(e, rj, acc[ii]);
    }
  }

  // Tail: j = iBase + t is valid only for rows ii > t (block-uniform t).
  for (int t = 0; t < ITILE - 1; ++t) {
    const int j = iBase + t;
    if (!rep_mask[b * SEQ + j]) continue;
    const float dj = dep[colBase + (size_t)j * DIM];
    const float rj = rep[colBase + (size_t)j * DIM];
#pragma unroll
    for (int ii = 1; ii < ITILE; ++ii) {
      if (ii > t) {
        const float tt = tanh_fast(0.2f * (dj + h[ii]));
        const float e  = __expf(5.0f * tt);
        ssum[ii] += e;
        acc[ii] = fmaf(e, rj, acc[ii]);
      }
    }
  }

#pragma unroll
  for (int ii = 0; ii < ITILE; ++ii) {
    out[((size_t)(b * SEQ + iBase + ii)) * DIM + d] =
        (ssum[ii] > 0.0f) ? (acc[ii] / ssum[ii]) : 0.0f;
  }
}

// ---------------------------------------------------------------------------
// Fused gate: G = sigmoid(attn @ Wa + rep @ Wr + f_bias)
//             out = (G*rep + (1-G)*attn) * rep_mask
// Single accumulator over a K=512 virtual reduction (two LDS panel passes).
// ---------------------------------------------------------------------------
__global__ __launch_bounds__(256) void fusion_wmma_kernel(
    const float* __restrict__ attn, const float* __restrict__ rep,
    const float* __restrict__ Wa, const float* __restrict__ Wr,
    const float* __restrict__ fbias, const uint8_t* __restrict__ rep_mask,
    float* __restrict__ out) {
  __shared__ float sW[DIM * 16];
  const int lane  = threadIdx.x & 31;
  const int wave  = threadIdx.x >> 5;
  const int nBase = blockIdx.y * 16;
  const int mBase = blockIdx.x * 128 + wave * 16;

  const int m     = mBase + (lane & 15);
  const int khalf = (lane >> 4) << 1;
  const int n     = lane & 15;

  v8f acc = {};
  for (int pass = 0; pass < 2; ++pass) {
    const float* __restrict__ Ap = pass ? rep : attn;
    const float* __restrict__ Wp = pass ? Wr : Wa;
    if (pass) __syncthreads();  // all waves done reading previous panel
    for (int idx = threadIdx.x; idx < DIM * 16; idx += 256) {
      const int r = idx >> 4, c = idx & 15;
      sW[idx] = Wp[r * DIM + nBase + c];
    }
    __syncthreads();
#pragma unroll 4
    for (int k = 0; k < DIM; k += 4) {
      const int kb = k + khalf;
      v2f a = *(const v2f*)(Ap + (size_t)m * DIM + kb);
      v2f b;
      b.x = sW[kb * 16 + n];
      b.y = sW[(kb + 1) * 16 + n];
      acc = __builtin_amdgcn_wmma_f32_16x16x4_f32(false, a, false, b, (short)0,
                                                  acc, false, false);
    }
  }

  const int mRow = mBase + (lane >> 4) * 8;
  const int col  = nBase + n;
#pragma unroll
  for (int r = 0; r < 8; ++r) {
    const int row = mRow + r;
    const float g   = 1.0f / (1.0f + __expf(-(acc[r] + fbias[col])));
    const size_t off = (size_t)row * DIM + col;
    const float o = g * rep[off] + (1.0f - g) * attn[off];
    out[off] = rep_mask[row] ? o : 0.0f;
  }
}

// ---------------------------------------------------------------------------
// Host launcher. Input dict order:
// 0 inputs[B,S,D] 1 rep_mask[B,S](bool) 2 rep_map_kernel 3 rep_map_bias
// 4 head_kernel 5 dependent_kernel 6 attn_bias 7 f_rep_kernel 8 f_attn_kernel
// 9 f_bias
// ---------------------------------------------------------------------------
extern "C" void kernel_launch(void* const* d_in, const int* in_sizes, int n_in,
                              void* d_out, int out_size, void* d_ws,
                              size_t ws_size, hipStream_t stream) {
  const float*   inputs    = (const float*)d_in[0];
  const uint8_t* rep_mask  = (const uint8_t*)d_in[1];
  const float*   W_rep     = (const float*)d_in[2];
  const float*   b_rep     = (const float*)d_in[3];
  const float*   W_head    = (const float*)d_in[4];
  const float*   W_dep     = (const float*)d_in[5];
  const float*   attn_bias = (const float*)d_in[6];
  const float*   W_frep    = (const float*)d_in[7];
  const float*   W_fattn   = (const float*)d_in[8];
  const float*   f_bias    = (const float*)d_in[9];

  const int M = in_sizes[0] / DIM;  // B*S = 3072
  const int B = M / SEQ;            // 8

  float* rep  = (float*)d_ws;                 // [M, DIM]
  float* dep  = rep  + (size_t)M * DIM;       // [M, DIM]
  float* head = dep  + (size_t)M * DIM;       // [M, DIM]
  float* attn = head + (size_t)M * DIM;       // [M, DIM]

  const dim3 gblk(256);
  const dim3 ggrid(M / 128, DIM / 16);

  hipLaunchKernelGGL((gemm_wmma_kernel<EPI_BIAS_ELU>), ggrid, gblk, 0, stream,
                     inputs, W_rep, b_rep, rep);
  hipLaunchKernelGGL((gemm_wmma_kernel<EPI_NONE>), ggrid, gblk, 0, stream,
                     rep, W_dep, (const float*)nullptr, dep);
  hipLaunchKernelGGL((gemm_wmma_kernel<EPI_NONE>), ggrid, gblk, 0, stream,
                     rep, W_head, (const float*)nullptr, head);

  hipLaunchKernelGGL(attn_softmax_kernel, dim3(SEQ / ITILE, B), dim3(DIM), 0,
                     stream, rep, dep, head, attn_bias, rep_mask, attn);

  hipLaunchKernelGGL(fusion_wmma_kernel, ggrid, gblk, 0, stream,
                     attn, rep, W_fattn, W_frep, f_bias, rep_mask,
                     (float*)d_out);
}